// RGCNRegressor_91268055040644
// MI455X (gfx1250) — compile-verified
//
#include <hip/hip_runtime.h>

// Problem constants (match reference)
#define NA_  131072
#define NW_  131072
#define NN_  262144          // NA + NW
#define EE_  1048576
#define RR_  4
#define DIN_ 64
#define HH_  128
#define LL_  2
#define EPS_ 1e-5f

typedef __attribute__((ext_vector_type(2))) float v2f;
typedef __attribute__((ext_vector_type(8))) float v8f;

// ---------------------------------------------------------------------------
// Generic 128-wide GEMM using V_WMMA_F32_16X16X4_F32 (fp32, exact vs reference)
//   out[rows,128]  (op)=  diag(scale?) * A[rows,K] @ W[K,128]  (+bias) (+out)
// Block = 256 threads = 8 wave32; each wave computes a 16x128 strip.
//
// W is staged in LDS in a pair-interleaved, bank-swizzled layout:
//   W[k][c] -> lws[(k/2)*256 + ((2c + ((k/2)&1)*32) & 255) + (k&1)]
// so each lane's B fragment {W[k][col], W[k+1][col]} is one contiguous
// ds_load_b64 into an even-aligned VGPR pair (direct WMMA src1, no movs),
// and the (p&1)*32 rotation makes the full-wave b64 access hit all 64 LDS
// banks exactly once (lanes 0-15 vs 16-31 use disjoint bank halves).
// All blocks are full -> EXEC all ones as WMMA requires.
// ---------------------------------------------------------------------------
template <int K, bool SCALE, bool BIAS, bool ACC, bool RELU>
__global__ __launch_bounds__(256) void gemm_k(const float* __restrict__ A,
                                              const float* __restrict__ cnt,
                                              const float* __restrict__ W,
                                              const float* __restrict__ bias,
                                              float* __restrict__ out)
{
    __shared__ float lws[K * HH_];   // K*128 floats, <= 64KB

    // cooperative swizzled store of W into LDS (float4 global reads)
    for (int idx = threadIdx.x; idx < K * (HH_ / 4); idx += 256) {
        float4 wv = reinterpret_cast<const float4*>(W)[idx];
        int kk = idx >> 5;            // row 0..K-1
        int c0 = (idx & 31) * 4;      // col base
        int p  = kk >> 1;
        int base = p * 256 + (kk & 1);
        int sw = (p & 1) * 32;
        lws[base + ((2 * (c0 + 0) + sw) & 255)] = wv.x;
        lws[base + ((2 * (c0 + 1) + sw) & 255)] = wv.y;
        lws[base + ((2 * (c0 + 2) + sw) & 255)] = wv.z;
        lws[base + ((2 * (c0 + 3) + sw) & 255)] = wv.w;
    }
    __syncthreads();

    const int wave = threadIdx.x >> 5;
    const int lane = threadIdx.x & 31;
    const int m    = lane & 15;       // row within 16-row tile / col within 16-col tile
    const int hi   = lane >> 4;       // lane-group: selects K pair / M half
    const int row0 = blockIdx.x * 128 + wave * 16;
    const int arow = row0 + m;

    float scale = 1.0f;
    if constexpr (SCALE) scale = 1.0f / fmaxf(cnt[arow], 1.0f);

    // Per-lane swizzled B offsets within a pair-row (dwords), one per col tile.
    // Note: p = k0/2 + hi and k0/2 is always even, so (p&1)*32 == hi*32.
    int boff[8];
#pragma unroll
    for (int ct = 0; ct < 8; ct++)
        boff[ct] = (ct * 32 + 2 * m + hi * 32) & 255;

    // C/D layout: acc[ct][i] = out[row0 + i + 8*hi][ct*16 + m]
    v8f acc[8];
#pragma unroll
    for (int ct = 0; ct < 8; ct++) {
        if constexpr (ACC) {
#pragma unroll
            for (int i = 0; i < 8; i++)
                acc[ct][i] = out[(size_t)(row0 + i + 8 * hi) * HH_ + ct * 16 + m];
        } else if constexpr (BIAS) {
            float bv = bias[ct * 16 + m];
#pragma unroll
            for (int i = 0; i < 8; i++) acc[ct][i] = bv;
        } else {
#pragma unroll
            for (int i = 0; i < 8; i++) acc[ct][i] = 0.0f;
        }
    }

    // K loop: 4 columns of A / rows of W per WMMA
    for (int k0 = 0; k0 < K; k0 += 4) {
        // A fragment: a[j] = A[arow][k0 + 2*hi + j]  (ISA 16x4 f32 layout)
        const float* ap = A + (size_t)arow * K + k0 + 2 * hi;
        v2f a;
        a.x = ap[0] * scale;
        a.y = ap[1] * scale;
        const float* lrow = &lws[(size_t)((k0 >> 1) + hi) * 256];
#pragma unroll
        for (int ct = 0; ct < 8; ct++) {
            // B fragment: b[j] = W[k0 + 2*hi + j][ct*16 + m]  (one ds_load_b64)
            v2f b = *reinterpret_cast<const v2f*>(lrow + boff[ct]);
            acc[ct] = __builtin_amdgcn_wmma_f32_16x16x4_f32(
                false, a, false, b, (short)0, acc[ct], false, false);
        }
    }

#pragma unroll
    for (int ct = 0; ct < 8; ct++) {
#pragma unroll
        for (int i = 0; i < 8; i++) {
            float v = acc[ct][i];
            if constexpr (RELU) v = fmaxf(v, 0.0f);
            out[(size_t)(row0 + i + 8 * hi) * HH_ + ct * 16 + m] = v;
        }
    }
}

// ---------------------------------------------------------------------------
// Zero fill (float4 grid-stride)
// ---------------------------------------------------------------------------
__global__ void zero_k(float* __restrict__ p, long n4)
{
    long i = (long)blockIdx.x * blockDim.x + threadIdx.x;
    long stride = (long)gridDim.x * blockDim.x;
    float4 z = make_float4(0.f, 0.f, 0.f, 0.f);
    for (; i < n4; i += stride) reinterpret_cast<float4*>(p)[i] = z;
}

// ---------------------------------------------------------------------------
// Edge scatter for one relation: one wave32 per edge.
//   agg[dst] += h[src]; cnt[dst] += 1   (only for edges of type r)
// Relation test is wave-uniform -> clean scalar branch, coalesced 512B gather.
// ---------------------------------------------------------------------------
__global__ __launch_bounds__(256) void scatter_k(const float* __restrict__ h,
                                                 const int* __restrict__ src,
                                                 const int* __restrict__ dst,
                                                 const int* __restrict__ etyp,
                                                 int r,
                                                 float* __restrict__ agg,
                                                 float* __restrict__ cnt)
{
    int e    = (int)((blockIdx.x * 256 + threadIdx.x) >> 5);
    int lane = threadIdx.x & 31;
    if (e >= EE_) return;
    if (etyp[e] != r) return;
    int s = src[e];
    int d = dst[e];
    float4 v = reinterpret_cast<const float4*>(h + (size_t)s * HH_)[lane];
    float* ad = agg + (size_t)d * HH_ + lane * 4;
    atomicAdd(ad + 0, v.x);
    atomicAdd(ad + 1, v.y);
    atomicAdd(ad + 2, v.z);
    atomicAdd(ad + 3, v.w);
    if (lane == 0) atomicAdd(cnt + d, 1.0f);
}

// ---------------------------------------------------------------------------
// LayerNorm (population var) + affine + ReLU: one wave32 per row of 128.
// ---------------------------------------------------------------------------
__global__ __launch_bounds__(256) void ln_relu_k(const float* __restrict__ x,
                                                 const float* __restrict__ g,
                                                 const float* __restrict__ b,
                                                 float* __restrict__ h,
                                                 int nrows)
{
    int row  = (int)((blockIdx.x * 256 + threadIdx.x) >> 5);
    int lane = threadIdx.x & 31;
    if (row >= nrows) return;

    float4 v = reinterpret_cast<const float4*>(x + (size_t)row * HH_)[lane];
    float s = v.x + v.y + v.z + v.w;
#pragma unroll
    for (int msk = 16; msk >= 1; msk >>= 1) s += __shfl_xor(s, msk, 32);
    float mu = s * (1.0f / HH_);

    float dx = v.x - mu, dy = v.y - mu, dz = v.z - mu, dw = v.w - mu;
    float q = dx * dx + dy * dy + dz * dz + dw * dw;
#pragma unroll
    for (int msk = 16; msk >= 1; msk >>= 1) q += __shfl_xor(q, msk, 32);
    float rstd = rsqrtf(q * (1.0f / HH_) + EPS_);

    int c = lane * 4;
    float4 o;
    o.x = fmaxf(dx * rstd * g[c + 0] + b[c + 0], 0.0f);
    o.y = fmaxf(dy * rstd * g[c + 1] + b[c + 1], 0.0f);
    o.z = fmaxf(dz * rstd * g[c + 2] + b[c + 2], 0.0f);
    o.w = fmaxf(dw * rstd * g[c + 3] + b[c + 3], 0.0f);
    reinterpret_cast<float4*>(h + (size_t)row * HH_)[lane] = o;
}

// ---------------------------------------------------------------------------
// Final head: pred[i] = base + h[i,:] . W_out + b_out. One wave32 per row.
// ---------------------------------------------------------------------------
__global__ __launch_bounds__(256) void pred_k(const float* __restrict__ h,
                                              const float* __restrict__ Wo,
                                              const float* __restrict__ bo,
                                              const float* __restrict__ base,
                                              float* __restrict__ pred,
                                              int nrows)
{
    int row  = (int)((blockIdx.x * 256 + threadIdx.x) >> 5);
    int lane = threadIdx.x & 31;
    if (row >= nrows) return;

    float4 v = reinterpret_cast<const float4*>(h + (size_t)row * HH_)[lane];
    float4 w = reinterpret_cast<const float4*>(Wo)[lane];
    float s = v.x * w.x + v.y * w.y + v.z * w.z + v.w * w.w;
#pragma unroll
    for (int msk = 16; msk >= 1; msk >>= 1) s += __shfl_xor(s, msk, 32);
    if (lane == 0) pred[row] = s + bo[0] + base[0];
}

// ---------------------------------------------------------------------------
// Host orchestration (graph-capture safe: only kernel launches on `stream`)
// ---------------------------------------------------------------------------
extern "C" void kernel_launch(void* const* d_in, const int* in_sizes, int n_in,
                              void* d_out, int out_size, void* d_ws, size_t ws_size,
                              hipStream_t stream)
{
    const float* x_a    = (const float*)d_in[0];
    const float* x_w    = (const float*)d_in[1];
    const int*   eidx   = (const int*)d_in[2];
    const int*   etyp   = (const int*)d_in[3];
    const float* W_in_a = (const float*)d_in[4];
    const float* b_in_a = (const float*)d_in[5];
    const float* W_in_w = (const float*)d_in[6];
    const float* b_in_w = (const float*)d_in[7];
    const float* W_rel  = (const float*)d_in[8];   // [L,R,H,H]
    const float* W_root = (const float*)d_in[9];   // [L,H,H]
    const float* b_conv = (const float*)d_in[10];  // [L,H]
    const float* ln_g   = (const float*)d_in[11];  // [L,H]
    const float* ln_b   = (const float*)d_in[12];  // [L,H]
    const float* W_out  = (const float*)d_in[13];  // [H,1]
    const float* b_out  = (const float*)d_in[14];  // [1]
    const float* base   = (const float*)d_in[15];  // [1]

    // Workspace carve-up: h | out | agg | cnt
    float* h   = (float*)d_ws;
    float* ob  = h   + (size_t)NN_ * HH_;
    float* agg = ob  + (size_t)NN_ * HH_;
    float* cnt = agg + (size_t)NN_ * HH_;

    const int* src = eidx;         // edge_index[0]
    const int* dst = eidx + EE_;   // edge_index[1]

    // Input projections: h = relu(x @ W_in + b_in)
    gemm_k<DIN_, false, true, false, true>
        <<<NA_ / 128, 256, 0, stream>>>(x_a, nullptr, W_in_a, b_in_a, h);
    gemm_k<DIN_, false, true, false, true>
        <<<NW_ / 128, 256, 0, stream>>>(x_w, nullptr, W_in_w, b_in_w,
                                        h + (size_t)NA_ * HH_);

    for (int l = 0; l < LL_; l++) {
        // out = h @ W_root[l] + b_conv[l]
        gemm_k<HH_, false, true, false, false>
            <<<NN_ / 128, 256, 0, stream>>>(h, nullptr,
                                            W_root + (size_t)l * HH_ * HH_,
                                            b_conv + l * HH_, ob);
        for (int r = 0; r < RR_; r++) {
            zero_k<<<2048, 256, 0, stream>>>(agg, (long)NN_ * HH_ / 4);
            zero_k<<<64, 256, 0, stream>>>(cnt, (long)NN_ / 4);
            scatter_k<<<EE_ / 8, 256, 0, stream>>>(h, src, dst, etyp, r, agg, cnt);
            // out += (agg / max(cnt,1)) @ W_rel[l,r]
            gemm_k<HH_, true, false, true, false>
                <<<NN_ / 128, 256, 0, stream>>>(agg, cnt,
                                                W_rel + ((size_t)l * RR_ + r) * HH_ * HH_,
                                                nullptr, ob);
        }
        // h = relu(layernorm(out))
        ln_relu_k<<<NN_ / 8, 256, 0, stream>>>(ob, ln_g + l * HH_, ln_b + l * HH_,
                                               h, NN_);
    }

    // pred = base + h[:NA] @ W_out + b_out
    pred_k<<<NA_ / 8, 256, 0, stream>>>(h, W_out, b_out, base, (float*)d_out, NA_);
}